// GCN_65335042506944
// MI455X (gfx1250) — compile-verified
//
#include <hip/hip_runtime.h>
#include <hip/hip_bf16.h>

// GCN forward on MI455X (gfx1250, wave32).
// GEMM1: fp32 WMMA fed from LDS, double-buffered via GLOBAL_LOAD_ASYNC_TO_LDS_B128.
// SpMM:  L2-resident gathers + global_atomic_add_f32 scatters.
// GEMM2: fp32 WMMA with fused bias+ReLU on the A operand.
// Tail:  fused +b2 log_softmax, one wave32 per row.

#define N_NODES 100000
#define N_EDGES 3200000
#define NFEAT   512
#define NHID    256
#define NCLASS  40

typedef __attribute__((ext_vector_type(2))) float v2f;
typedef __attribute__((ext_vector_type(8))) float v8f;

// Async copy 16B from global to LDS (ASYNCcnt-tracked, no VGPR round-trip).
__device__ __forceinline__ void async_ld_b128(unsigned lds_byte_off, const float* gaddr) {
    asm volatile("global_load_async_to_lds_b128 %0, %1, off"
                 :
                 : "v"(lds_byte_off), "v"(gaddr)
                 : "memory");
}
__device__ __forceinline__ void wait_async0() {
    asm volatile("s_wait_asynccnt 0x0" ::: "memory");
}

// ---------------------------------------------------------------------------
// GEMM1: support[100000 x 256] = x[100000 x 512] @ W1[512 x 256]
// Block = 8 waves covering a 16(M) x 128(N) output strip.
// K-chunks of 32 staged in LDS: A 16x32 (2KB) + B 32x128 (16KB), double buffered.
// ---------------------------------------------------------------------------
#define KC     32                      // K-chunk
#define NCHUNK (NFEAT / KC)            // 16 chunks
#define A_FLT  (16 * KC)               // 512 floats
#define B_FLT  (KC * 128)              // 4096 floats
#define BUF_FLT (A_FLT + B_FLT)        // 4608 floats per buffer

__global__ void gemm1_wmma(const float* __restrict__ x,
                           const float* __restrict__ W1,
                           float* __restrict__ support) {
    __shared__ float lds[2 * BUF_FLT];         // 36 KB

    const int t    = threadIdx.x;              // 0..255
    const int lane = t & 31;
    const int wave = t >> 5;                   // 0..7 -> 16-col subtile
    const int half = lane >> 4;                // 0 | 1
    const int l15  = lane & 15;

    const int bm      = blockIdx.x >> 1;       // 6250 row-tiles
    const int cg      = blockIdx.x & 1;        // 2 col-groups of 128
    const int rowBase = bm * 16;
    const int colBase = cg * 128;
    const int colLoc  = wave * 16 + l15;       // 0..127 within strip

    const unsigned ldsBase = (unsigned)(unsigned long long)(&lds[0]); // LDS byte offset

    // ---- issue async loads for chunk ci into buffer ci&1 --------------------
    auto issue_chunk = [&](int ci) {
        const int k0 = ci * KC;
        const unsigned bufB = ldsBase + (unsigned)((ci & 1) * BUF_FLT) * 4u;
        // B: 32x128 floats = 1024 vec4; thread t covers q = t, t+256, t+512, t+768
#pragma unroll
        for (int j = 0; j < 4; ++j) {
            const int q  = t + j * 256;
            const int kk = q >> 5;             // 0..31
            const int c4 = (q & 31) << 2;      // 0,4,..,124
            async_ld_b128(bufB + (unsigned)(A_FLT * 4 + q * 16),
                          W1 + (size_t)(k0 + kk) * NHID + colBase + c4);
        }
        // A: 16x32 floats = 128 vec4; threads 0..127
        if (t < 128) {
            const int row = t >> 3;            // 0..15
            const int c4  = (t & 7) << 2;      // 0..28
            async_ld_b128(bufB + (unsigned)(t * 16),
                          x + (size_t)(rowBase + row) * NFEAT + k0 + c4);
        }
    };

    issue_chunk(0);

    v8f c = {0.f, 0.f, 0.f, 0.f, 0.f, 0.f, 0.f, 0.f};

    for (int ci = 0; ci < NCHUNK; ++ci) {
        wait_async0();                 // my async stores to LDS landed
        __syncthreads();               // everyone's landed; prev buffer free
        if (ci + 1 < NCHUNK) issue_chunk(ci + 1);

        const float* bufA = &lds[(ci & 1) * BUF_FLT];
        const float* bufB = bufA + A_FLT;

#pragma unroll
        for (int kk = 0; kk < KC; kk += 4) {
            // A frag: lanes 0-15 hold K=kk..kk+1, lanes 16-31 hold K=kk+2..kk+3
            v2f a = *(const v2f*)&bufA[l15 * KC + kk + half * 2];   // ds b64
            // B frag: V0 = rows {K0|K2}, V1 = rows {K1|K3}
            v2f b;
            b.x = bufB[(kk + half * 2) * 128 + colLoc];             // ds b32
            b.y = bufB[(kk + half * 2 + 1) * 128 + colLoc];
            c = __builtin_amdgcn_wmma_f32_16x16x4_f32(
                    false, a, false, b, (short)0, c, false, false);
        }
    }

    // D layout: VGPR r -> row (r | r+8); lanes cover 16 cols
    float* dptr = support + (size_t)(rowBase + half * 8) * NHID + colBase + colLoc;
#pragma unroll
    for (int r = 0; r < 8; ++r)
        dptr[(size_t)r * NHID] = c[r];
}

// ---------------------------------------------------------------------------
// SpMM1: h[row] += val * support[col]  (256 feats; h pre-zeroed)
// Block = 256 threads (one per feature), 16 edges per block.
// ---------------------------------------------------------------------------
#define EPB1 16
__global__ void spmm1_scatter(const int* __restrict__ erow,
                              const int* __restrict__ ecol,
                              const float* __restrict__ eval,
                              const float* __restrict__ support,
                              float* __restrict__ h) {
    const int t = threadIdx.x;                      // feature 0..255
    const size_t eBase = (size_t)blockIdx.x * EPB1;
#pragma unroll 2
    for (int i = 0; i < EPB1; ++i) {
        const size_t e = eBase + i;                 // N_EDGES % EPB1 == 0
        const int r = erow[e];
        const int c = ecol[e];
        const float v = eval[e];
        if (i + 1 < EPB1) {                         // warm next gather row in L2
            const int cn = ecol[e + 1];
            __builtin_prefetch(&support[(size_t)cn * NHID + t], 0, 3);
        }
        const float g = support[(size_t)c * NHID + t] * v;
        unsafeAtomicAdd(&h[(size_t)r * NHID + t], g);
    }
}

// ---------------------------------------------------------------------------
// GEMM2: t2[100000 x 40] = relu(h + b1) @ W2[256 x 40]  (N padded 40->48)
// ---------------------------------------------------------------------------
#define NTILES2 (6250 * 3)
__global__ void gemm2_wmma(const float* __restrict__ h,
                           const float* __restrict__ b1,
                           const float* __restrict__ W2,
                           float* __restrict__ t2) {
    const int lane = threadIdx.x & 31;
    const int wave = threadIdx.x >> 5;
    const int tile = blockIdx.x * 8 + wave;
    if (tile >= NTILES2) return;
    const int tileM = tile / 3;
    const int tileN = tile % 3;
    const int half = lane >> 4;
    const int l15  = lane & 15;

    const int rowA = tileM * 16 + l15;
    const int colB = tileN * 16 + l15;
    const bool colOK = (colB < NCLASS);

    const float* aptr = h + (size_t)rowA * NHID + half * 2;
    const float* bptr = W2 + (size_t)(half * 2) * NCLASS + colB;
    const float* bias = b1 + half * 2;

    v8f c = {0.f, 0.f, 0.f, 0.f, 0.f, 0.f, 0.f, 0.f};

#pragma unroll 4
    for (int k = 0; k < NHID; k += 4) {
        v2f a = *(const v2f*)aptr;
        a.x = fmaxf(a.x + bias[0], 0.f);            // fused +b1, ReLU
        a.y = fmaxf(a.y + bias[1], 0.f);
        v2f b;
        b.x = colOK ? bptr[0]      : 0.f;
        b.y = colOK ? bptr[NCLASS] : 0.f;
        c = __builtin_amdgcn_wmma_f32_16x16x4_f32(
                false, a, false, b, (short)0, c, false, false);
        aptr += 4;
        bptr += 4 * NCLASS;
        bias += 4;
    }

    if (colOK) {
        float* dptr = t2 + (size_t)(tileM * 16 + half * 8) * NCLASS + colB;
#pragma unroll
        for (int r = 0; r < 8; ++r)
            dptr[(size_t)r * NCLASS] = c[r];
    }
}

// ---------------------------------------------------------------------------
// SpMM2: out[row] += val * t2[col]  (40 feats; one wave32 per edge)
// ---------------------------------------------------------------------------
__global__ void spmm2_scatter(const int* __restrict__ erow,
                              const int* __restrict__ ecol,
                              const float* __restrict__ eval,
                              const float* __restrict__ t2,
                              float* __restrict__ out) {
    const int lane = threadIdx.x & 31;
    const size_t e = (size_t)blockIdx.x * 8 + (threadIdx.x >> 5);
    if (e >= N_EDGES) return;
    const int r = erow[e];
    const int c = ecol[e];
    const float v = eval[e];

    const float g0 = t2[(size_t)c * NCLASS + lane] * v;
    unsafeAtomicAdd(&out[(size_t)r * NCLASS + lane], g0);
    if (lane < NCLASS - 32) {
        const float g1 = t2[(size_t)c * NCLASS + 32 + lane] * v;
        unsafeAtomicAdd(&out[(size_t)r * NCLASS + 32 + lane], g1);
    }
}

// ---------------------------------------------------------------------------
// Fused +b2 and row-wise log_softmax over 40 classes, in place in d_out.
// ---------------------------------------------------------------------------
__global__ void logsoftmax_rows(float* __restrict__ out,
                                const float* __restrict__ b2) {
    const int lane = threadIdx.x & 31;
    const int row = blockIdx.x * 8 + (threadIdx.x >> 5);
    if (row >= N_NODES) return;
    float* rp = out + (size_t)row * NCLASS;

    float v0 = rp[lane] + b2[lane];
    float v1 = (lane < 8) ? (rp[32 + lane] + b2[32 + lane]) : -3.0e38f;

    float m = fmaxf(v0, v1);
#pragma unroll
    for (int off = 16; off >= 1; off >>= 1)
        m = fmaxf(m, __shfl_xor(m, off, 32));

    float s = __expf(v0 - m) + ((lane < 8) ? __expf(v1 - m) : 0.f);
#pragma unroll
    for (int off = 16; off >= 1; off >>= 1)
        s += __shfl_xor(s, off, 32);

    const float lse = m + __logf(s);
    rp[lane] = v0 - lse;
    if (lane < 8) rp[32 + lane] = v1 - lse;
}

// ---------------------------------------------------------------------------
extern "C" void kernel_launch(void* const* d_in, const int* in_sizes, int n_in,
                              void* d_out, int out_size, void* d_ws, size_t ws_size,
                              hipStream_t stream) {
    const float* x    = (const float*)d_in[0];
    const int*   erow = (const int*)  d_in[1];
    const int*   ecol = (const int*)  d_in[2];
    const float* evl  = (const float*)d_in[3];
    const float* W1   = (const float*)d_in[4];
    const float* b1   = (const float*)d_in[5];
    const float* W2   = (const float*)d_in[6];
    const float* b2   = (const float*)d_in[7];
    float* out = (float*)d_out;

    // Workspace: [support 102.4MB][h 102.4MB]; t2 reuses the support region
    float* support = (float*)d_ws;
    float* h       = support + (size_t)N_NODES * NHID;
    float* t2      = support;   // support dead after spmm1; safe reuse

    // GEMM1: 6250 row-tiles x 2 col-groups -> 12500 blocks of 8 waves
    gemm1_wmma<<<12500, 256, 0, stream>>>(x, W1, support);

    hipMemsetAsync(h, 0, (size_t)N_NODES * NHID * sizeof(float), stream);
    spmm1_scatter<<<N_EDGES / EPB1, 256, 0, stream>>>(erow, ecol, evl, support, h);

    gemm2_wmma<<<(NTILES2 + 7) / 8, 256, 0, stream>>>(h, b1, W2, t2);

    hipMemsetAsync(out, 0, (size_t)N_NODES * NCLASS * sizeof(float), stream);
    spmm2_scatter<<<N_EDGES / 8, 256, 0, stream>>>(erow, ecol, evl, t2, out);

    logsoftmax_rows<<<N_NODES / 8, 256, 0, stream>>>(out, b2);
}